// MiniBatchDiscrimination_87720412053850
// MI455X (gfx1250) — compile-verified
//
#include <hip/hip_runtime.h>

typedef __attribute__((ext_vector_type(2))) float v2f;
typedef __attribute__((ext_vector_type(8))) float v8f;

#define BSZ   1024   // batch
#define FDIM  2048   // in features
#define KKER  50     // kernels
#define DDIM  5      // dims per kernel
#define KD    250    // K*D
#define OUTW  2098   // F + K

// ---------------------------------------------------------------------------
// Stage 1: act^T[c][i] = sum_f x[i][f] * W[f][c] + bias[c]
// f32 WMMA 16x16x4. One wave per block; each wave: one 16-row M tile times
// four 16-col N tiles (4 accumulators, shared A load per k-step).
// A layout (16x4 f32): lane l: M = l%16, VGPR0/1 hold K = 2*(l/16) + {0,1}.
// B layout (4x16 f32): lane l: N = l%16, VGPR0/1 hold K = 2*(l/16) + {0,1}.
// D layout (16x16 f32): lane l, VGPR v: M = v + 8*(l/16), N = l%16.
// ---------------------------------------------------------------------------
__global__ __launch_bounds__(32) void gemm_wmma_f32(const float* __restrict__ x,
                                                    const float* __restrict__ w,
                                                    const float* __restrict__ bias,
                                                    float* __restrict__ actT) {
  const int lane = threadIdx.x & 31;
  const int half = lane >> 4;   // 0 or 1 (selects K pair)
  const int l16  = lane & 15;
  const int i0 = blockIdx.x * 16;   // M tile base (rows of x)
  const int c0 = blockIdx.y * 64;   // N group base (4 tiles of 16 cols)

  v8f acc[4] = {v8f{}, v8f{}, v8f{}, v8f{}};

  // clamp OOB columns (N padded 250 -> 256); garbage lanes never stored
  int c_t[4];
#pragma unroll
  for (int t = 0; t < 4; ++t) {
    int c = c0 + t * 16 + l16;
    c_t[t] = (c < KD) ? c : (KD - 1);
  }

  const float* xrow = x + (size_t)(i0 + l16) * FDIM + 2 * half;

  for (int kk = 0; kk < FDIM; kk += 4) {
    v2f a = *(const v2f*)(xrow + kk);            // b64 load, 8B aligned
    const int krow = kk + 2 * half;
    const float* wb0 = w + (size_t)krow * KD;    // row K
    const float* wb1 = wb0 + KD;                 // row K+1
#pragma unroll
    for (int t = 0; t < 4; ++t) {
      v2f b;
      b.x = wb0[c_t[t]];
      b.y = wb1[c_t[t]];
      acc[t] = __builtin_amdgcn_wmma_f32_16x16x4_f32(
          /*neg_a=*/false, a, /*neg_b=*/false, b,
          /*c_mod=*/(short)0, acc[t], /*reuse_a=*/false, /*reuse_b=*/false);
    }
  }

  // store transposed: actT[c * 1024 + m], add bias
#pragma unroll
  for (int t = 0; t < 4; ++t) {
    int c = c0 + t * 16 + l16;
    if (c < KD) {
      float bv = bias[c];
#pragma unroll
      for (int v = 0; v < 8; ++v) {
        int m = i0 + v + 8 * half;
        actT[(size_t)c * BSZ + m] = acc[t][v] + bv;
      }
    }
  }
}

// ---------------------------------------------------------------------------
// Stage 2: per kernel k, LDS-resident act[:,k,:] (5 x 1024 f32 = 20 KB).
// Each thread owns one row i; loops j over LDS (broadcast reads, no bank
// conflicts). minibatch[i,k] = sum_j exp(-(L1(i,j) + (i==j))).
// ---------------------------------------------------------------------------
__global__ __launch_bounds__(256) void pairwise_l1(const float* __restrict__ actT,
                                                   float* __restrict__ out) {
  __shared__ float sa[DDIM * BSZ];
  const int k = blockIdx.x;                         // 0..49
  const int i = blockIdx.y * 256 + threadIdx.x;     // 0..1023

  // rows k*5 .. k*5+4 of actT are contiguous: one flat coalesced copy
  const float* base = actT + (size_t)k * DDIM * BSZ;
  for (int idx = threadIdx.x; idx < DDIM * BSZ; idx += 256)
    sa[idx] = base[idx];
  __syncthreads();

  const float a0 = sa[i];
  const float a1 = sa[BSZ + i];
  const float a2 = sa[2 * BSZ + i];
  const float a3 = sa[3 * BSZ + i];
  const float a4 = sa[4 * BSZ + i];

  float acc = 0.0f;
#pragma unroll 4
  for (int j = 0; j < BSZ; ++j) {
    float d = __builtin_fabsf(a0 - sa[j])
            + __builtin_fabsf(a1 - sa[BSZ + j])
            + __builtin_fabsf(a2 - sa[2 * BSZ + j])
            + __builtin_fabsf(a3 - sa[3 * BSZ + j])
            + __builtin_fabsf(a4 - sa[4 * BSZ + j]);
    d += (j == i) ? 1.0f : 0.0f;                    // eps = eye(B)
    acc += __expf(-d);                              // v_exp_f32
  }

  out[(size_t)i * OUTW + FDIM + k] = acc;
}

// ---------------------------------------------------------------------------
// Stage 3: out[:, 0:2048] = x  (float2 keeps 8B alignment since 2098 is even)
// ---------------------------------------------------------------------------
__global__ __launch_bounds__(256) void copy_x(const float* __restrict__ x,
                                              float* __restrict__ out) {
  int t = blockIdx.x * 256 + threadIdx.x;  // 0 .. 1024*1024-1 (float2 units)
  int i  = t >> 10;
  int c2 = t & 1023;
  float2 v = ((const float2*)x)[t];
  *(float2*)(out + (size_t)i * OUTW + c2 * 2) = v;
}

extern "C" void kernel_launch(void* const* d_in, const int* in_sizes, int n_in,
                              void* d_out, int out_size, void* d_ws, size_t ws_size,
                              hipStream_t stream) {
  const float* x    = (const float*)d_in[0];  // [1024, 2048]
  const float* w    = (const float*)d_in[1];  // [2048, 250]
  const float* bias = (const float*)d_in[2];  // [250]
  float* out  = (float*)d_out;                // [1024, 2098]
  float* actT = (float*)d_ws;                 // [250, 1024] scratch (~1 MB)

  copy_x<<<dim3((BSZ * FDIM / 2) / 256), dim3(256), 0, stream>>>(x, out);
  gemm_wmma_f32<<<dim3(BSZ / 16, 4), dim3(32), 0, stream>>>(x, w, bias, actT);
  pairwise_l1<<<dim3(KKER, BSZ / 256), dim3(256), 0, stream>>>(actT, out);
}